// DET_PROB_9964324127180
// MI455X (gfx1250) — compile-verified
//
#include <hip/hip_runtime.h>

// DET_PROB hierarchy: B0=8, B1=16, B2=16, BATCH=32768.
// Memory-bound streaming kernel (~554 MB total @ 23.3 TB/s => ~24 us floor).
// CDNA5 path used: GLOBAL_LOAD_ASYNC_TO_LDS_B128 (th:TH_LOAD_NT) double-
// buffered prefetch tracked with ASYNCcnt, segmented product scans via
// wave32 shuffles, nontemporal b128 output stores.

#define DP_B0 8
#define DP_ROWLEN 2048   // B0*B1*B2 floats per row
#define DP_DC1LEN 128    // B0*B1
#define DP_TPB 256
#define DP_NBLK 4096

typedef float v4f __attribute__((ext_vector_type(4)));

__global__ __launch_bounds__(DP_TPB) void det_prob_kernel(
    const float* __restrict__ dc0,
    const float* __restrict__ dc1,
    const float* __restrict__ dc2,
    float* __restrict__ out,
    int B)
{
    // Double-buffered dc2 row staging (2 x 8KB) + group-prefix broadcast.
    __shared__ __align__(16) float sdc2[2][DP_ROWLEN];
    __shared__ float sPref[DP_DC1LEN];

    const int t    = (int)threadIdx.x;
    const int step = (int)gridDim.x;

    // Block-relative LDS byte offset of this thread's first 16B chunk.
    // (generic pointer to __shared__: low 32 bits == LDS byte offset)
    const unsigned ldsBase =
        (unsigned)(size_t)(&sdc2[0][0]) + (unsigned)(t * 16);

    int row = blockIdx.x;          // launch clamps grid so blockIdx.x < B
    int cur = 0;

    // Prologue: async-prefetch first row into buffer 0 (streamed once ->
    // non-temporal so the 286MB read stream doesn't churn the 192MB L2).
    {
        const float* g = dc2 + (size_t)row * DP_ROWLEN + (size_t)(t * 4);
        asm volatile("global_load_async_to_lds_b128 %0, %1, off th:TH_LOAD_NT"
                     :: "v"(ldsBase), "v"(g) : "memory");
        asm volatile("global_load_async_to_lds_b128 %0, %1, off offset:4096 th:TH_LOAD_NT"
                     :: "v"(ldsBase), "v"(g) : "memory");
    }

    while (true) {
        const int  nextrow  = row + step;
        const bool havenext = nextrow < B;

        // Issue prefetch of next row into the other buffer (overlaps with
        // the prefix computation and this row's compute below).
        if (havenext) {
            const unsigned lp = ldsBase + ((unsigned)(cur ^ 1) << 13); // +8192B
            const float* g = dc2 + (size_t)nextrow * DP_ROWLEN + (size_t)(t * 4);
            asm volatile("global_load_async_to_lds_b128 %0, %1, off th:TH_LOAD_NT"
                         :: "v"(lp), "v"(g) : "memory");
            asm volatile("global_load_async_to_lds_b128 %0, %1, off offset:4096 th:TH_LOAD_NT"
                         :: "v"(lp), "v"(g) : "memory");
        }

        // ---- Group prefix: prefix[g] = cumprod(dc0)[g/16] * seg16-scan(dc1)[g]
        if (t < DP_DC1LEN) {                    // waves 0..3, fully active
            const int a0 = t >> 4;
            const float* p0 = dc0 + (size_t)row * DP_B0;
            float c0v = 1.0f;
            for (int j = 0; j <= a0; ++j) c0v *= p0[j];   // <=8 cached loads

            float s = dc1[(size_t)row * DP_DC1LEN + t];
            const int p16 = t & 15;
            #pragma unroll
            for (int d = 1; d < 16; d <<= 1) {
                float o = __shfl_up(s, (unsigned)d, 16);
                if (p16 >= d) s *= o;
            }
            sPref[t] = c0v * s;
        }
        __syncthreads();

        // Wait for this row's async loads. In-order async completion means
        // cnt<=2 leaves only the just-issued next-row pair outstanding.
        if (havenext) asm volatile("s_wait_asynccnt 2" ::: "memory");
        else          asm volatile("s_wait_asynccnt 0" ::: "memory");

        // ---- Compute: each thread owns one 4-float chunk of a 16-group,
        // twice per row. 4-lane segmented product scan stitches chunks.
        const int lane4 = t & 3;
        float* orow = out + (size_t)row * DP_ROWLEN;
        #pragma unroll
        for (int r = 0; r < 2; ++r) {
            const float pfx = sPref[(r << 6) + (t >> 2)];
            const v4f v = *(const v4f*)&sdc2[cur][r * 1024 + t * 4];
            const float m0 = v.x;
            const float m1 = m0 * v.y;
            const float m2 = m1 * v.z;
            const float m3 = m2 * v.w;

            float sg = m3;                       // chunk total
            float o  = __shfl_up(sg, 1u, 4); if (lane4 >= 1) sg *= o;
            o        = __shfl_up(sg, 2u, 4); if (lane4 >= 2) sg *= o;
            float e  = __shfl_up(sg, 1u, 4); if (lane4 == 0) e = 1.0f;

            const float pe = pfx * e;
            v4f ov;
            ov.x = pe * m0; ov.y = pe * m1; ov.z = pe * m2; ov.w = pe * m3;
            // write-once 268MB stream: keep it out of L2's working set
            __builtin_nontemporal_store(ov, (v4f*)(orow + r * 1024 + t * 4));
        }
        __syncthreads();    // protect sPref WAR for next iteration

        if (!havenext) break;
        row = nextrow;
        cur ^= 1;
    }
}

extern "C" void kernel_launch(void* const* d_in, const int* in_sizes, int n_in,
                              void* d_out, int out_size, void* d_ws, size_t ws_size,
                              hipStream_t stream) {
    (void)n_in; (void)out_size; (void)d_ws; (void)ws_size;
    const float* dc0 = (const float*)d_in[0];
    const float* dc1 = (const float*)d_in[1];
    const float* dc2 = (const float*)d_in[2];
    float* out = (float*)d_out;

    const int B = in_sizes[0] / DP_B0;     // batch rows
    int nblk = DP_NBLK;
    if (nblk > B) nblk = B;

    det_prob_kernel<<<nblk, DP_TPB, 0, stream>>>(dc0, dc1, dc2, out, B);
}